// DAS_dual_12309376270528
// MI455X (gfx1250) — compile-verified
//
#include <hip/hip_runtime.h>
#include <hip/hip_bf16.h>
#include <stdint.h>

// DAS dual-speed-of-sound beamforming for MI455X (gfx1250).
//
// Shape constants from the reference:
#define N_TR      256          // transducers
#define T_SAMP    2048         // time samples per transducer row
#define NPIX      65536        // 256x256 output pixels
#define INV_TS    (1.0f / 2.5e-8f)
#define CHUNKS    16           // transducer chunks (partial planes in d_ws)
#define NT_CHUNK  (N_TR / CHUNKS)   // 16 transducers per chunk
#define TILE_PIX  2048         // pixels per block
#define TILES     (NPIX / TILE_PIX) // 32 pixel tiles
#define PPT       (TILE_PIX / 256)  // 8 pixels per thread

// ---- CDNA5 async global->LDS copy (ASYNCcnt path) -------------------------
// Builtin signature (from hipcc diagnostic): param 1 is v4i32 in AS1 (global),
// param 2 is the LDS-side pointer (AS3), then imm offset + imm cpol.
typedef int v4i __attribute__((ext_vector_type(4)));
typedef __attribute__((address_space(1))) v4i gv4i_t;
typedef __attribute__((address_space(3))) v4i lv4i_t;

__device__ __forceinline__ void async_copy_b128(const float* g, float* l) {
#if defined(__HIP_DEVICE_COMPILE__) && __has_builtin(__builtin_amdgcn_global_load_async_to_lds_b128)
    // 16B per lane, memory -> LDS, tracked by ASYNCcnt (no VGPR data path)
    __builtin_amdgcn_global_load_async_to_lds_b128((gv4i_t*)g, (lv4i_t*)l, 0, 0);
#else
    // Fallback: synchronous copy through VGPRs (global_load_b128 + ds_store_b128)
    *(float4*)l = *(const float4*)g;
#endif
}

__device__ __forceinline__ void wait_async_all() {
#if defined(__HIP_DEVICE_COMPILE__)
#if __has_builtin(__builtin_amdgcn_s_wait_asynccnt)
    __builtin_amdgcn_s_wait_asynccnt(0);
#else
    asm volatile("s_wait_asynccnt 0" ::: "memory");
#endif
#endif
}

// ---- Pass 1: partial sums over a chunk of transducers ----------------------
// grid = (TILES, CHUNKS), block = 256 threads (8 wave32).
// Each block: for 16 transducers, DMA the 8KB sinogram row into LDS
// (double-buffered), stream geometry with coalesced b128 loads, gather from
// LDS, accumulate 8 pixels/thread in registers.
__global__ __launch_bounds__(256)
void das_partial_kernel(const float* __restrict__ sino,
                        const float* __restrict__ v0p,
                        const float* __restrict__ v1p,
                        const float* __restrict__ ddp,
                        const float* __restrict__ rep,
                        const float* __restrict__ dtx,
                        const float* __restrict__ dby,
                        float* __restrict__ partial)
{
    __shared__ __align__(16) float row[2][T_SAMP];   // 16 KB double buffer

    const int tid     = threadIdx.x;
    const int tile    = blockIdx.x;
    const int chunk   = blockIdx.y;
    const int pixBase = tile * TILE_PIX + tid * PPT;
    const int n0      = chunk * NT_CHUNK;

    // Uniform scalars (1-element device arrays)
    const float rv0 = 1.0f / v0p[0];
    const float rv1 = 1.0f / v1p[0];
    const float off = rep[0] - ddp[0];
    // idx = rint( ((a - b + off)*rv0 + b*rv1) * INV_TS )

    float acc[PPT];
#pragma unroll
    for (int j = 0; j < PPT; ++j) acc[j] = 0.0f;

    // Prefetch first row into buffer 0: 2048 floats = 256 lanes x 2 x b128
    {
        const float* g = sino + (size_t)n0 * T_SAMP;
        async_copy_b128(g + tid * 4,          &row[0][tid * 4]);
        async_copy_b128(g + (tid + 256) * 4,  &row[0][(tid + 256) * 4]);
    }
    wait_async_all();
    __syncthreads();

    for (int k = 0; k < NT_CHUNK; ++k) {
        const int n = n0 + k;
        const float* rcur = row[k & 1];

        // Prefetch next row into the other buffer; overlaps with compute
        // below, fenced by s_wait_asynccnt + barrier at loop bottom.
        if (k + 1 < NT_CHUNK) {
            const float* g = sino + (size_t)(n + 1) * T_SAMP;
            float* l = row[(k + 1) & 1];
            async_copy_b128(g + tid * 4,         l + tid * 4);
            async_copy_b128(g + (tid + 256) * 4, l + (tid + 256) * 4);
        }

        // Coalesced streaming of geometry (the bandwidth-critical 128 MB)
        const float4* pa = (const float4*)(dtx + (size_t)n * NPIX + pixBase);
        const float4* pb = (const float4*)(dby + (size_t)n * NPIX + pixBase);
#pragma unroll
        for (int q = 0; q < PPT / 4; ++q) {
            float4 a4 = pa[q];
            float4 b4 = pb[q];
            float av[4] = { a4.x, a4.y, a4.z, a4.w };
            float bv[4] = { b4.x, b4.y, b4.z, b4.w };
#pragma unroll
            for (int e = 0; e < 4; ++e) {
                float f = ((av[e] - bv[e] + off) * rv0 + bv[e] * rv1) * INV_TS;
                int   i = (int)__builtin_rintf(f);   // round-half-even == jnp.round
                // Reference zeroes samples 0 and T-1 then clips -> only
                // indices 1..T-2 can contribute.
                float s = (i >= 1 && i <= T_SAMP - 2) ? rcur[i] : 0.0f;
                acc[q * 4 + e] += s;
            }
        }

        wait_async_all();   // my prefetch slice landed in LDS
        __syncthreads();    // everyone done reading rcur + everyone's slice visible
    }

    float* o = partial + (size_t)chunk * NPIX + pixBase;
#pragma unroll
    for (int q = 0; q < PPT / 4; ++q) {
        float4 v;
        v.x = acc[q * 4 + 0];
        v.y = acc[q * 4 + 1];
        v.z = acc[q * 4 + 2];
        v.w = acc[q * 4 + 3];
        ((float4*)o)[q] = v;
    }
}

// ---- Pass 2: deterministic reduction of the 16 partial planes --------------
__global__ __launch_bounds__(256)
void das_reduce_kernel(const float* __restrict__ partial,
                       float* __restrict__ out)
{
    const int pix = blockIdx.x * 256 + threadIdx.x;
    float s = 0.0f;
#pragma unroll
    for (int c = 0; c < CHUNKS; ++c)
        s += partial[(size_t)c * NPIX + pix];
    out[pix] = s * (1.0f / (float)N_TR);
}

extern "C" void kernel_launch(void* const* d_in, const int* in_sizes, int n_in,
                              void* d_out, int out_size, void* d_ws, size_t ws_size,
                              hipStream_t stream) {
    // setup_inputs() order:
    //   0: sinogram [256,2048] f32      1: v0 (scalar)   2: v1 (scalar)
    //   3: d_delay (scalar)             4: ring_error (scalar)
    //   5: dist_tx [256,256,256] f32    6: dist_body [256,256,256] f32
    const float* sino = (const float*)d_in[0];
    const float* v0   = (const float*)d_in[1];
    const float* v1   = (const float*)d_in[2];
    const float* dd   = (const float*)d_in[3];
    const float* re   = (const float*)d_in[4];
    const float* dtx  = (const float*)d_in[5];
    const float* dby  = (const float*)d_in[6];

    float* partial = (float*)d_ws;   // needs CHUNKS*NPIX*4 = 4 MB scratch
    float* out     = (float*)d_out;  // 65536 f32

    das_partial_kernel<<<dim3(TILES, CHUNKS), 256, 0, stream>>>(
        sino, v0, v1, dd, re, dtx, dby, partial);
    das_reduce_kernel<<<NPIX / 256, 256, 0, stream>>>(partial, out);
}